// kernel_generated_2_62904091017431
// MI455X (gfx1250) — compile-verified
//
#include <hip/hip_runtime.h>

// D = A×B + C via V_WMMA_F32_16X16X4_F32 (CDNA5, wave32).
// A: 16x4 f32  -> 2 VGPRs/lane (v2f). Lanes 0-15: M=lane, K={kk,kk+1}; lanes 16-31: M=lane-16, K={kk+2,kk+3}.
// B: 4x16 f32  -> 2 VGPRs/lane (v2f). Lanes 0-15: N=lane, K={kk,kk+1}; lanes 16-31: N=lane-16, K={kk+2,kk+3}.
// C/D: 16x16 f32 -> 8 VGPRs/lane (v8f). VGPR r: lanes 0-15 M=r, lanes 16-31 M=r+8; N=lane&15.
typedef float v2f __attribute__((ext_vector_type(2)));
typedef float v8f __attribute__((ext_vector_type(8)));

#define C_CH   128
#define H_DIM  14
#define W_DIM  14
#define HW     (H_DIM * W_DIM)     // 196
#define J_OUT  32
#define KTOT   384                 // 3 taps * 128 channels
#define LDSTR  386                 // padded W row stride (even -> 8B aligned b64 reads)

__global__ __launch_bounds__(256)
void conv3tap_wmma_f32(const float* __restrict__ x,
                       const float* __restrict__ Wt,
                       float* __restrict__ out) {
  __shared__ float w_lds[J_OUT * LDSTR];   // 49.4 KB of the 320 KB WGP LDS

  const int tid = threadIdx.x;

  // Stage W (32 x 384 f32, layout j*384 + k*128 + i == GEMM K order) into padded LDS.
  for (int t = tid; t < J_OUT * KTOT; t += 256) {
    const int j = t / KTOT;
    const int K = t - j * KTOT;
    w_lds[j * LDSTR + K] = Wt[t];
  }
  __syncthreads();

  const int wave = tid >> 5;          // 8 waves per block
  const int lane = tid & 31;
  const int half = lane >> 4;         // 0: lanes 0-15, 1: lanes 16-31
  const int lq   = lane & 15;         // N index within the 16-wide spatial tile
  const int koff = half << 1;         // K sub-offset {0,2} for this lane half

  // Spatial column handled by this lane: s = n*196 + h*14 + w (output coords).
  const int tile = blockIdx.x * 8 + wave;          // 1568 tiles total
  const int s    = tile * 16 + lq;
  const int w    = s % W_DIM;
  const int h    = (s / W_DIM) % H_DIM;
  const int n    = s / HW;
  const int hs   = (h + H_DIM - 1) % H_DIM;        // output roll along h: read y at h-1

  v8f c0 = {};   // j = 0..15
  v8f c1 = {};   // j = 16..31

  // A fragment base pointers in LDS (per-lane row, K offset folded in).
  const float* wl0 = &w_lds[(0  + lq) * LDSTR + koff];
  const float* wl1 = &w_lds[(16 + lq) * LDSTR + koff];

  // Warm the B gather stream (global_prefetch_b8): first element of this lane's column.
  __builtin_prefetch(x + ((size_t)n * C_CH * H_DIM + hs) * W_DIM, 0, 0);

  for (int k = 0; k < 3; ++k) {
    // tap k reads x at width (w+k-2) mod 14, masked to 0 outside 0 <= w+k-1 < 14.
    const int  wsrc  = (w + k + W_DIM - 2) % W_DIM;
    const int  wk    = w + k - 1;
    const bool valid = (wk >= 0) && (wk < W_DIM);
    // Branch-free boundary mask: the address is always legal (wsrc in range);
    // only the VALUE is zeroed. One v_cndmask per tap, then v_mul in the loop
    // (exact: finite * 0.0f == 0.0f). This keeps the global loads unconditional
    // so no EXEC-predicated branches appear in the K-loop.
    const float bscale = valid ? 1.0f : 0.0f;
    // &x[n, i=koff, hs, wsrc]; channel i advances in strides of HW floats
    const float* bp = x + (size_t)n * C_CH * HW + (size_t)koff * HW
                        + (size_t)hs * W_DIM + wsrc;

#pragma unroll 8
    for (int ib = 0; ib < C_CH; ib += 4) {          // 32 K-steps per tap, 96 total
      const int kk = k * C_CH + ib;                 // GEMM K base for this step
      v2f a0, a1, b;
      a0.x = wl0[kk];     a0.y = wl0[kk + 1];       // ds_load_b64 (even offsets)
      a1.x = wl1[kk];     a1.y = wl1[kk + 1];
      const float* bpp = bp + (size_t)ib * HW;      // immediate offsets after unroll
      b.x = bpp[0]  * bscale;                       // channel i = ib + koff
      b.y = bpp[HW] * bscale;                       // channel i = ib + koff + 1
      // v_wmma_f32_16x16x4_f32 : D = A*B + C
      c0 = __builtin_amdgcn_wmma_f32_16x16x4_f32(false, a0, false, b,
                                                 (short)0, c0, false, false);
      c1 = __builtin_amdgcn_wmma_f32_16x16x4_f32(false, a1, false, b,
                                                 (short)0, c1, false, false);
    }
  }

  // Scatter D: VGPR r holds (M = r + 8*half) at column N = lq -> out[n, j, h, w].
  const int m0 = half * 8;
  float* op = out + ((size_t)n * J_OUT * H_DIM + h) * W_DIM + w;
#pragma unroll
  for (int r = 0; r < 8; ++r) {
    const int j = m0 + r;
    op[(size_t)j        * HW] = c0[r];
    op[(size_t)(16 + j) * HW] = c1[r];
  }
}

extern "C" void kernel_launch(void* const* d_in, const int* in_sizes, int n_in,
                              void* d_out, int out_size, void* d_ws, size_t ws_size,
                              hipStream_t stream) {
  const float* x  = (const float*)d_in[0];   // (128,128,14,14) f32
  const float* Wt = (const float*)d_in[1];   // (32,3,128) f32
  float* out = (float*)d_out;                // (128,32,14,14) f32
  (void)in_sizes; (void)n_in; (void)out_size; (void)d_ws; (void)ws_size;
  // 1568 spatial tiles of 16, 8 waves (tiles) per 256-thread block -> 196 blocks
  hipLaunchKernelGGL(conv3tap_wmma_f32, dim3(196), dim3(256), 0, stream, x, Wt, out);
}